// GraphLayer_57449482551584
// MI455X (gfx1250) — compile-verified
//
#include <hip/hip_runtime.h>
#include <hip/hip_bf16.h>

typedef int          v4i  __attribute__((ext_vector_type(4)));
typedef float        v4f  __attribute__((ext_vector_type(4)));
typedef unsigned int u32x4 __attribute__((ext_vector_type(4)));
typedef int          i32x4 __attribute__((ext_vector_type(4)));
typedef int          i32x8 __attribute__((ext_vector_type(8)));

#define TILE 1024          // edges per block (one wave), 12KB LDS
#define EPL  (TILE / 32)   // edges per lane (contiguous -> long runs)

// ---------------------------------------------------------------------------
// Kernel 1: zero the Az accumulator in workspace (must be re-zeroed per call)
// ---------------------------------------------------------------------------
__global__ void az_zero_kernel(float* __restrict__ Az, int n) {
    int i4 = (blockIdx.x * blockDim.x + threadIdx.x) * 4;
    if (i4 + 4 <= n) {
        v4f zero = {0.f, 0.f, 0.f, 0.f};
        *(v4f*)(Az + i4) = zero;
    } else {
        for (int i = i4; i < n; ++i) Az[i] = 0.f;
    }
}

// ---------------------------------------------------------------------------
// TDM: issue a 1-D tensor_load_to_lds of `nelem` 4-byte elements starting at
// `gptr` into LDS byte offset `lds_off`. D# layout per CDNA5 ISA §8.3-8.5.
// ---------------------------------------------------------------------------
__device__ __forceinline__ void tdm_load_1d(unsigned lds_off, const void* gptr,
                                            int nelem) {
    unsigned long long ga = (unsigned long long)gptr;

    // Group 0: count=1 | lds_addr | global_addr[56:0] | type=2 (bits 127:126)
    u32x4 g0;
    g0[0] = 1u;                                   // count=1, no gather, user D#
    g0[1] = lds_off;                              // LDS byte address
    g0[2] = (unsigned)ga;                         // global_addr[31:0]
    g0[3] = (unsigned)((ga >> 32) & 0x01FFFFFFu)  // global_addr[56:32]
            | 0x80000000u;                        // type=2 ("image")

    // Group 1: data_size=4B (code 2); tensor_dim0 = tile_dim0 = nelem; 1-D.
    i32x8 g1;
    g1[0] = (2 << 16);                            // data_size=4B
    g1[1] = (int)((nelem & 0xFFFF) << 16);        // tensor_dim0[15:0]  @bits 63:48
    g1[2] = (nelem >> 16) & 0xFFFF;               // tensor_dim0[31:16] @bits 79:64
    g1[3] = (int)((nelem & 0xFFFF) << 16);        // tile_dim0 @bits 127:112
    g1[4] = 0;                                    // tile_dim1=0 (1-D), tile_dim2=0
    g1[5] = nelem;                                // tensor_dim0_stride (unused, 1-D)
    g1[6] = 0;
    g1[7] = 0;

    i32x4 zero4 = {0, 0, 0, 0};
    i32x8 zero8 = {0, 0, 0, 0, 0, 0, 0, 0};
    __builtin_amdgcn_tensor_load_to_lds(g0, g1, zero4, zero4, zero8, 0);
}

// ---------------------------------------------------------------------------
// Kernel 2: Az[rows[e]] += edge_vals[e] * z[cols[e]]
// One wave per block. The wave DMAs its 1024-edge tile (rows/cols/vals) into
// LDS via the Tensor Data Mover, waits on TENSORcnt, then each lane
// run-length-compresses its 32 contiguous (sorted) rows before issuing
// global_atomic_add_f32. z (4MB) stays L2-resident; edge streams never touch
// a VGPR load slot.
// ---------------------------------------------------------------------------
__global__ void __launch_bounds__(32)
spmv_scatter_kernel(const float* __restrict__ z,
                    const float* __restrict__ ev,
                    const int*   __restrict__ rows,
                    const int*   __restrict__ cols,
                    float*       __restrict__ Az,
                    int n_edges) {
    __shared__ int   lds_r[TILE];
    __shared__ int   lds_c[TILE];
    __shared__ float lds_v[TILE];

    long long tile_base = (long long)blockIdx.x * TILE;
    if (tile_base >= n_edges) return;

    if (tile_base + TILE <= (long long)n_edges) {
        // --- TDM stage: 3 async tensor DMAs, tracked with TENSORcnt -------
        unsigned off_r = (unsigned)(unsigned long long)(uintptr_t)lds_r;
        unsigned off_c = (unsigned)(unsigned long long)(uintptr_t)lds_c;
        unsigned off_v = (unsigned)(unsigned long long)(uintptr_t)lds_v;

        tdm_load_1d(off_r, rows + tile_base, TILE);
        tdm_load_1d(off_c, cols + tile_base, TILE);
        tdm_load_1d(off_v, ev   + tile_base, TILE);

        __builtin_amdgcn_s_wait_tensorcnt(0);
        __builtin_amdgcn_fence(__ATOMIC_ACQUIRE, "workgroup");
        asm volatile("" ::: "memory");

        // --- Gather + run-compressed scatter ------------------------------
        const int lane = threadIdx.x;
        const int base = lane * EPL;           // contiguous slice -> long runs

        int   cur = -1;
        float acc = 0.0f;
#pragma unroll
        for (int chunk = 0; chunk < EPL; chunk += 8) {
            int   r[8];
            int   c[8];
            float v[8];
#pragma unroll
            for (int i = 0; i < 8; i += 4) {
                v4i rr = *(const v4i*)(lds_r + base + chunk + i);
                v4i cc = *(const v4i*)(lds_c + base + chunk + i);
                v4f vv = *(const v4f*)(lds_v + base + chunk + i);
                r[i + 0] = rr.x; r[i + 1] = rr.y; r[i + 2] = rr.z; r[i + 3] = rr.w;
                c[i + 0] = cc.x; c[i + 1] = cc.y; c[i + 2] = cc.z; c[i + 3] = cc.w;
                v[i + 0] = vv.x; v[i + 1] = vv.y; v[i + 2] = vv.z; v[i + 3] = vv.w;
            }
            // prefetch the gather targets (global_prefetch_b8) ahead of the
            // dependent sequential accumulation below
#pragma unroll
            for (int i = 0; i < 8; ++i) {
                __builtin_prefetch(z + c[i], 0, 3);
            }
#pragma unroll
            for (int i = 0; i < 8; ++i) {
                float prod = v[i] * z[c[i]];
                int   ri   = r[i];
                if (ri == cur) {
                    acc += prod;
                } else {
                    if (cur >= 0) atomicAdd(Az + cur, acc);
                    cur = ri;
                    acc = prod;
                }
            }
        }
        if (cur >= 0) atomicAdd(Az + cur, acc);
    } else {
        // scalar tail (strided across the wave)
        for (long long e = tile_base + threadIdx.x; e < (long long)n_edges; e += 32) {
            atomicAdd(Az + rows[e], ev[e] * z[cols[e]]);
        }
    }
}

// ---------------------------------------------------------------------------
// Kernel 3: out = leaky_relu(alpha*D^g*z + beta*D^(g-1)*Az + bias, slope)
// ---------------------------------------------------------------------------
__global__ void finalize_kernel(const float* __restrict__ z,
                                const float* __restrict__ D,
                                const float* __restrict__ Az,
                                const float* __restrict__ params,
                                float*       __restrict__ out,
                                int n) {
    const float p0 = params[0], p1 = params[1], p2 = params[2];
    const float p3 = params[3], p4 = params[4];

    const float alpha = __expf(p0);
    const float beta  = alpha * tanhf(p1);
    const float gamma = 1.0f / (1.0f + __expf(-p2));
    const float bias  = p3;
    const float slope = fmaxf(p4, 0.0f) + log1pf(__expf(-fabsf(p4)));  // softplus

    int i4 = (blockIdx.x * blockDim.x + threadIdx.x) * 4;
    if (i4 + 4 <= n) {
        v4f zz = *(const v4f*)(z  + i4);
        v4f dd = *(const v4f*)(D  + i4);
        v4f aa = *(const v4f*)(Az + i4);
        v4f oo;
#pragma unroll
        for (int k = 0; k < 4; ++k) {
            float d  = dd[k];
            float dg = exp2f(gamma * log2f(d));   // D^gamma (D in [1,64])
            float t  = alpha * dg * zz[k] + beta * (dg / d) * aa[k] + bias;
            oo[k] = (t >= 0.0f) ? t : slope * t;
        }
        __builtin_nontemporal_store(oo, (v4f*)(out + i4));
    } else {
        for (int i = i4; i < n; ++i) {
            float d  = D[i];
            float dg = exp2f(gamma * log2f(d));
            float t  = alpha * dg * z[i] + beta * (dg / d) * Az[i] + bias;
            out[i] = (t >= 0.0f) ? t : slope * t;
        }
    }
}

// ---------------------------------------------------------------------------
extern "C" void kernel_launch(void* const* d_in, const int* in_sizes, int n_in,
                              void* d_out, int out_size, void* d_ws, size_t ws_size,
                              hipStream_t stream) {
    const float* z      = (const float*)d_in[0];
    const float* params = (const float*)d_in[1];
    const float* D      = (const float*)d_in[2];
    const float* ev     = (const float*)d_in[3];
    const int*   rows   = (const int*)d_in[4];
    const int*   cols   = (const int*)d_in[5];
    float*       out    = (float*)d_out;

    const int n       = in_sizes[0];   // number of nodes
    const int n_edges = in_sizes[3];   // number of edges

    float* Az = (float*)d_ws;          // n floats of scratch

    // 1) zero accumulator
    {
        int vec_threads = (n + 3) / 4;
        int blocks = (vec_threads + 255) / 256;
        az_zero_kernel<<<blocks, 256, 0, stream>>>(Az, n);
    }

    // 2) TDM-staged edge scatter with run compression
    {
        long long tiles = ((long long)n_edges + TILE - 1) / TILE;
        spmv_scatter_kernel<<<(int)tiles, 32, 0, stream>>>(z, ev, rows, cols, Az,
                                                           n_edges);
    }

    // 3) elementwise finalize
    {
        int vec_threads = (n + 3) / 4;
        int blocks = (vec_threads + 255) / 256;
        finalize_kernel<<<blocks, 256, 0, stream>>>(z, D, Az, params, out, n);
    }
}